// OctVolSynth_9981503996139
// MI455X (gfx1250) — compile-verified
//
#include <hip/hip_runtime.h>
#include <stdint.h>

// ---------------------------------------------------------------------------
// OctVolSynth: label gather + scale, global min/max normalize, label binarize.
// Memory-bound (no matrix math). CDNA5 path: GLOBAL_LOAD_ASYNC_TO_LDS_B128
// double-buffered staging with s_wait_asynccnt, LDS-resident intensity table,
// non-temporal output stores (write stream must not thrash the 192MB L2).
// ---------------------------------------------------------------------------

#define THREADS 256
#define EPT 4                 // elements per thread (16B async b128 chunk)
#define TILE (THREADS * EPT)  // 1024 elements per tile
#define MAX_LAB 1024          // LDS capacity for intensity table (513 used)

typedef int   v4i __attribute__((ext_vector_type(4)));
typedef float v4f __attribute__((ext_vector_type(4)));
typedef __attribute__((address_space(1))) v4i* gvp;  // global v4i*
typedef __attribute__((address_space(3))) v4i* lvp;  // LDS v4i*

// Async copy of 16 bytes per lane: global -> LDS. Tracked on ASYNCcnt.
__device__ __forceinline__ void async_b128(const void* g, void* l) {
#if __has_builtin(__builtin_amdgcn_global_load_async_to_lds_b128)
  __builtin_amdgcn_global_load_async_to_lds_b128(
      (gvp)(uintptr_t)g, (lvp)l, 0, 0);
#else
  unsigned lds_off = (unsigned)(uintptr_t)l;           // flat low 32b == LDS offset
  unsigned long long ga = (unsigned long long)(uintptr_t)g;
  asm volatile("global_load_async_to_lds_b128 %0, %1, off"
               :: "v"(lds_off), "v"(ga) : "memory");
#endif
}

#if __has_builtin(__builtin_amdgcn_s_wait_asynccnt)
#define WAIT_ASYNC(n) __builtin_amdgcn_s_wait_asynccnt(n)
#else
#define WAIT_ASYNC(n) asm volatile("s_wait_asynccnt %0" :: "i"(n) : "memory")
#endif

// ---------------------------------------------------------------------------
__global__ void init_ws(unsigned* ws) {
  if (threadIdx.x == 0) {
    ws[0] = 0x7F800000u;  // +inf bits: min accumulator (all values >= 0)
    ws[1] = 0x00000000u;  // 0.0 bits : max accumulator
  }
}

// ---------------------------------------------------------------------------
__global__ void reduce_minmax(const int* __restrict__ labels,
                              const float* __restrict__ paren,
                              const float* __restrict__ inten,
                              int n_labels, long long N,
                              unsigned* __restrict__ ws) {
  __shared__ float s_int[MAX_LAB];
  __shared__ float s_p[2][TILE];
  __shared__ int   s_l[2][TILE];
  __shared__ float s_rmin[THREADS / 32];
  __shared__ float s_rmax[THREADS / 32];

  const int tid = threadIdx.x;
  for (int i = tid; i < n_labels && i < MAX_LAB; i += THREADS) s_int[i] = inten[i];
  __syncthreads();

  const long long tiles  = N / TILE;
  const long long stride = gridDim.x;
  float vmin = __uint_as_float(0x7F800000u);  // +inf
  float vmax = 0.0f;                          // values are >= 0

  long long t = blockIdx.x;
  int buf = 0;
  if (t < tiles) {
    long long b = t * TILE + (long long)tid * EPT;
    async_b128(paren + b,  &s_p[0][tid * EPT]);
    async_b128(labels + b, &s_l[0][tid * EPT]);
  }
  for (; t < tiles; t += stride) {
    long long nxt = t + stride;
    if (nxt < tiles) {  // prefetch next tile into the other buffer
      long long nb = nxt * TILE + (long long)tid * EPT;
      async_b128(paren + nb,  &s_p[buf ^ 1][tid * EPT]);
      async_b128(labels + nb, &s_l[buf ^ 1][tid * EPT]);
      WAIT_ASYNC(2);  // current tile's 2 async ops complete (in-order)
    } else {
      WAIT_ASYNC(0);
    }
    // Each thread consumes exactly the 16B it loaded -> no barrier needed.
    v4f p = *(const v4f*)&s_p[buf][tid * EPT];
    v4i l = *(const v4i*)&s_l[buf][tid * EPT];
    float v0 = p.x * (l.x == 0 ? 1.0f : (l.x < MAX_LAB ? s_int[l.x] : inten[l.x]));
    float v1 = p.y * (l.y == 0 ? 1.0f : (l.y < MAX_LAB ? s_int[l.y] : inten[l.y]));
    float v2 = p.z * (l.z == 0 ? 1.0f : (l.z < MAX_LAB ? s_int[l.z] : inten[l.z]));
    float v3 = p.w * (l.w == 0 ? 1.0f : (l.w < MAX_LAB ? s_int[l.w] : inten[l.w]));
    vmin = fminf(vmin, fminf(fminf(v0, v1), fminf(v2, v3)));
    vmax = fmaxf(vmax, fmaxf(fmaxf(v0, v1), fmaxf(v2, v3)));
    buf ^= 1;
  }

  // Tail (N % TILE), handled scalar by block 0.
  if (blockIdx.x == 0) {
    for (long long i = tiles * TILE + tid; i < N; i += THREADS) {
      int  l = labels[i];
      float v = paren[i] * (l == 0 ? 1.0f : (l < MAX_LAB ? s_int[l] : inten[l]));
      vmin = fminf(vmin, v);
      vmax = fmaxf(vmax, v);
    }
  }

  // Wave32 reduce, then cross-wave via LDS, then 2 atomics per block.
  for (int off = 16; off >= 1; off >>= 1) {
    vmin = fminf(vmin, __shfl_xor(vmin, off, 32));
    vmax = fmaxf(vmax, __shfl_xor(vmax, off, 32));
  }
  const int wid = tid >> 5, lane = tid & 31;
  if (lane == 0) { s_rmin[wid] = vmin; s_rmax[wid] = vmax; }
  __syncthreads();
  if (tid == 0) {
    float m = s_rmin[0], M = s_rmax[0];
#pragma unroll
    for (int i = 1; i < THREADS / 32; ++i) {
      m = fminf(m, s_rmin[i]);
      M = fmaxf(M, s_rmax[i]);
    }
    // values >= 0 -> fp32 bits order like unsigned ints
    atomicMin(&ws[0], __float_as_uint(m));
    atomicMax(&ws[1], __float_as_uint(M));
  }
}

// ---------------------------------------------------------------------------
__global__ void finalize(const int* __restrict__ labels,
                         const float* __restrict__ paren,
                         const float* __restrict__ inten,
                         int n_labels, long long N,
                         const unsigned* __restrict__ ws,
                         float* __restrict__ outf,
                         int* __restrict__ outi) {
  __shared__ float s_int[MAX_LAB];
  __shared__ float s_p[2][TILE];
  __shared__ int   s_l[2][TILE];

  const int tid = threadIdx.x;
  for (int i = tid; i < n_labels && i < MAX_LAB; i += THREADS) s_int[i] = inten[i];
  __syncthreads();

  const float mn    = __uint_as_float(ws[0]);
  const float mx    = __uint_as_float(ws[1]);
  const float denom = mx - mn;  // == max(final - min) exactly (monotone fp32 sub)

  const long long tiles  = N / TILE;
  const long long stride = gridDim.x;

  long long t = blockIdx.x;
  int buf = 0;
  if (t < tiles) {
    long long b = t * TILE + (long long)tid * EPT;
    async_b128(paren + b,  &s_p[0][tid * EPT]);
    async_b128(labels + b, &s_l[0][tid * EPT]);
  }
  for (; t < tiles; t += stride) {
    long long nxt = t + stride;
    if (nxt < tiles) {
      long long nb = nxt * TILE + (long long)tid * EPT;
      async_b128(paren + nb,  &s_p[buf ^ 1][tid * EPT]);
      async_b128(labels + nb, &s_l[buf ^ 1][tid * EPT]);
      WAIT_ASYNC(2);
    } else {
      WAIT_ASYNC(0);
    }
    v4f p = *(const v4f*)&s_p[buf][tid * EPT];
    v4i l = *(const v4i*)&s_l[buf][tid * EPT];
    v4f r;
    r.x = (p.x * (l.x == 0 ? 1.0f : (l.x < MAX_LAB ? s_int[l.x] : inten[l.x])) - mn) / denom;
    r.y = (p.y * (l.y == 0 ? 1.0f : (l.y < MAX_LAB ? s_int[l.y] : inten[l.y])) - mn) / denom;
    r.z = (p.z * (l.z == 0 ? 1.0f : (l.z < MAX_LAB ? s_int[l.z] : inten[l.z])) - mn) / denom;
    r.w = (p.w * (l.w == 0 ? 1.0f : (l.w < MAX_LAB ? s_int[l.w] : inten[l.w])) - mn) / denom;
    v4i b4;
    b4.x = (l.x >= 1) ? 1 : l.x;
    b4.y = (l.y >= 1) ? 1 : l.y;
    b4.z = (l.z >= 1) ? 1 : l.z;
    b4.w = (l.w >= 1) ? 1 : l.w;
    long long o = t * TILE + (long long)tid * EPT;
    // Output is write-once streaming data: keep it out of L2 (NT stores).
    __builtin_nontemporal_store(r,  (v4f*)(outf + o));
    __builtin_nontemporal_store(b4, (v4i*)(outi + o));
    buf ^= 1;
  }

  if (blockIdx.x == 0) {
    for (long long i = tiles * TILE + tid; i < N; i += THREADS) {
      int l = labels[i];
      float v = paren[i] * (l == 0 ? 1.0f : (l < MAX_LAB ? s_int[l] : inten[l]));
      __builtin_nontemporal_store((v - mn) / denom, outf + i);
      __builtin_nontemporal_store((l >= 1) ? 1 : l, outi + i);
    }
  }
}

// ---------------------------------------------------------------------------
extern "C" void kernel_launch(void* const* d_in, const int* in_sizes, int n_in,
                              void* d_out, int out_size, void* d_ws, size_t ws_size,
                              hipStream_t stream) {
  const int*   labels = (const int*)d_in[0];
  const float* paren  = (const float*)d_in[1];
  const float* inten  = (const float*)d_in[2];
  const long long N   = (long long)in_sizes[0];
  const int n_labels  = in_sizes[2];

  unsigned* ws   = (unsigned*)d_ws;
  float*    outf = (float*)d_out;
  int*      outi = (int*)((float*)d_out + N);  // outputs concatenated: final, binary

  long long tiles = N / TILE;
  int blocks = (tiles <= 0) ? 1 : (int)(tiles < 2048 ? tiles : 2048);

  init_ws<<<1, 32, 0, stream>>>(ws);
  reduce_minmax<<<blocks, THREADS, 0, stream>>>(labels, paren, inten, n_labels, N, ws);
  finalize<<<blocks, THREADS, 0, stream>>>(labels, paren, inten, n_labels, N, ws,
                                           outf, outi);
}